// HeterogeneousGraphSAGE_78752520339773
// MI455X (gfx1250) — compile-verified
//
#include <hip/hip_runtime.h>

#define DFEAT 128
#define SCAN_B 256

typedef float v2f __attribute__((ext_vector_type(2)));
typedef float v8f __attribute__((ext_vector_type(8)));

// ---------------- zero int array ----------------
__global__ void sage_zero_int_kernel(int* __restrict__ p, int n) {
    int i = blockIdx.x * blockDim.x + threadIdx.x;
    int stride = gridDim.x * blockDim.x;
    for (; i < n; i += stride) p[i] = 0;
}

// ---------------- histogram: cnt[dst[e]]++ ----------------
__global__ void sage_hist_kernel(const int* __restrict__ dst,
                                 int* __restrict__ cnt, int E) {
    int e = blockIdx.x * blockDim.x + threadIdx.x;
    if (e < E) atomicAdd(&cnt[dst[e]], 1);
}

// ---------------- per-block exclusive scan (Hillis-Steele) ----------------
__global__ __launch_bounds__(SCAN_B)
void sage_scan_block_kernel(const int* __restrict__ in, int* __restrict__ offs,
                            int* __restrict__ blockSums, int n) {
    __shared__ int tmp[SCAN_B];
    const int tid = threadIdx.x;
    const int gid = blockIdx.x * SCAN_B + tid;
    int v = (gid < n) ? in[gid] : 0;
    tmp[tid] = v;
    __syncthreads();
    #pragma unroll
    for (int off = 1; off < SCAN_B; off <<= 1) {
        int t = (tid >= off) ? tmp[tid - off] : 0;
        __syncthreads();
        tmp[tid] += t;
        __syncthreads();
    }
    if (gid < n) offs[gid] = tmp[tid] - v;            // exclusive
    if (tid == SCAN_B - 1) blockSums[blockIdx.x] = tmp[SCAN_B - 1];
}

// ---------------- scan block sums (tiny, runs once, single thread) -------
__global__ void sage_scan_sums_kernel(int* __restrict__ sums, int nb) {
    if (blockIdx.x == 0 && threadIdx.x == 0) {
        int run = 0;
        for (int i = 0; i < nb; ++i) { int t = sums[i]; sums[i] = run; run += t; }
    }
}

// ---------------- add block offsets; copy to cursor; seal offs[N] --------
__global__ __launch_bounds__(SCAN_B)
void sage_scan_add_kernel(int* __restrict__ offs, const int* __restrict__ blockSums,
                          int* __restrict__ cursor, int n, int E) {
    int gid = blockIdx.x * SCAN_B + threadIdx.x;
    if (gid < n) {
        int v = offs[gid] + blockSums[blockIdx.x];
        offs[gid]   = v;
        cursor[gid] = v;
    }
    if (gid == 0) offs[n] = E;
}

// ---------------- bucket fill: CSR edge list sorted by dst ----------------
__global__ void sage_fill_kernel(const int* __restrict__ src,
                                 const int* __restrict__ dst,
                                 int* __restrict__ cursor,
                                 int* __restrict__ eSorted, int E) {
    int e = blockIdx.x * blockDim.x + threadIdx.x;
    if (e < E) {
        int p = atomicAdd(&cursor[dst[e]], 1);
        eSorted[p] = src[e];
    }
}

// ---------------- gather-mean aggregation: zero atomics ------------------
// one wave32 per node; lane owns 4 contiguous floats of the 128-float row.
// Streams neighbor rows (512B coalesced, L2-resident) into register
// accumulators; global_prefetch 4 rows ahead hides the eSorted->row chase.
__global__ __launch_bounds__(256)
void sage_aggregate_kernel(const float* __restrict__ feat,
                           const int* __restrict__ offs,
                           const int* __restrict__ eSorted,
                           float* __restrict__ agg, int N) {
    int node = blockIdx.x * (blockDim.x >> 5) + (threadIdx.x >> 5);
    int lane = threadIdx.x & 31;
    if (node >= N) return;
    const int beg = offs[node];
    const int end = offs[node + 1];

    float4 a0 = make_float4(0.f, 0.f, 0.f, 0.f);
    float4 a1 = make_float4(0.f, 0.f, 0.f, 0.f);
    int j = beg;
    for (; j + 1 < end; j += 2) {                    // 2-deep for MLP
        int s0 = eSorted[j];
        int s1 = eSorted[j + 1];
        if (j + 5 < end) {                           // prefetch 4 edges ahead
            int sp0 = eSorted[j + 4];
            int sp1 = eSorted[j + 5];
            __builtin_prefetch(feat + (long long)sp0 * DFEAT + lane * 4, 0, 0);
            __builtin_prefetch(feat + (long long)sp1 * DFEAT + lane * 4, 0, 0);
        }
        const float4 v0 = *(const float4*)(feat + (long long)s0 * DFEAT + lane * 4);
        const float4 v1 = *(const float4*)(feat + (long long)s1 * DFEAT + lane * 4);
        a0.x += v0.x; a0.y += v0.y; a0.z += v0.z; a0.w += v0.w;
        a1.x += v1.x; a1.y += v1.y; a1.z += v1.z; a1.w += v1.w;
    }
    if (j < end) {
        int s0 = eSorted[j];
        const float4 v0 = *(const float4*)(feat + (long long)s0 * DFEAT + lane * 4);
        a0.x += v0.x; a0.y += v0.y; a0.z += v0.z; a0.w += v0.w;
    }
    a0.x += a1.x; a0.y += a1.y; a0.z += a1.z; a0.w += a1.w;

    const float inv = (end > beg) ? 1.0f / (float)(end - beg) : 1.0f; // max(deg,1)
    a0.x *= inv; a0.y *= inv; a0.z *= inv; a0.w *= inv;
    *(float4*)(agg + (long long)node * DFEAT + lane * 4) = a0;
}

// ---------------- fused SAGE layer GEMM via V_WMMA_F32_16X16X4_F32 -------
// out[i,:] = relu( agg[i,:] @ Wl + bl + X[i,:] @ Wr )   (agg already mean)
// block = 256 threads = 8 waves; block owns 16 rows; wave w owns cols [16w,16w+16)
// Tiles staged with GLOBAL_LOAD_ASYNC_TO_LDS_B128 (ASYNCcnt path, no VGPR
// round-trip). X and out may alias (layer 2): X tile fully staged to LDS
// before any store, row-tiles disjoint across blocks.
__global__ __launch_bounds__(256)
void sage_gemm_kernel(const float* __restrict__ agg,
                      const float* X,
                      const float* __restrict__ Wl,
                      const float* __restrict__ bl,
                      const float* __restrict__ Wr,
                      float* out, int N) {
    __shared__ float sA[16][132];   // pad 132: m-stride -> 4 banks apart
    __shared__ float sX[16][132];

    const int tid = threadIdx.x;
    const int rowBase = blockIdx.x * 16;

    const float4 z4 = make_float4(0.f, 0.f, 0.f, 0.f);
    #pragma unroll
    for (int i = 0; i < 2; ++i) {
        int idx = tid + i * 256;        // float4 index in [0,512)
        int r   = idx >> 5;             // 32 float4 per row
        int c4  = idx & 31;
        if ((rowBase + r) < N) {
            // direct global -> LDS DMA (tracked by ASYNCcnt)
            unsigned ldsA = (unsigned)(unsigned long long)&sA[r][c4 * 4];
            unsigned ldsX = (unsigned)(unsigned long long)&sX[r][c4 * 4];
            const float* ga = agg + (long long)(rowBase + r) * DFEAT + c4 * 4;
            const float* gx = X   + (long long)(rowBase + r) * DFEAT + c4 * 4;
            asm volatile("global_load_async_to_lds_b128 %0, %1, off"
                         :: "v"(ldsA), "v"(ga) : "memory");
            asm volatile("global_load_async_to_lds_b128 %0, %1, off"
                         :: "v"(ldsX), "v"(gx) : "memory");
        } else {
            *(float4*)&sA[r][c4 * 4] = z4;
            *(float4*)&sX[r][c4 * 4] = z4;
        }
    }
    asm volatile("s_wait_asynccnt 0x0" ::: "memory");
    __syncthreads();

    const int wave = tid >> 5;
    const int lane = tid & 31;
    const int m    = lane & 15;        // A: M-row / B,C,D: N-col within tile
    const int hi   = lane >> 4;        // selects K pair {0,1} vs {2,3}
    const int colBase = wave * 16;

    v8f c = {};
    #pragma unroll 8
    for (int kt = 0; kt < 32; ++kt) {
        int k = kt * 4 + 2 * hi;
        v2f a1, b1, a2, b2;
        a1.x = sA[m][k];
        a1.y = sA[m][k + 1];
        b1.x = Wl[k * DFEAT + colBase + m];
        b1.y = Wl[(k + 1) * DFEAT + colBase + m];
        c = __builtin_amdgcn_wmma_f32_16x16x4_f32(false, a1, false, b1,
                                                  (short)0, c, false, false);
        a2.x = sX[m][k];
        a2.y = sX[m][k + 1];
        b2.x = Wr[k * DFEAT + colBase + m];
        b2.y = Wr[(k + 1) * DFEAT + colBase + m];
        c = __builtin_amdgcn_wmma_f32_16x16x4_f32(false, a2, false, b2,
                                                  (short)0, c, false, false);
    }

    const float bb = bl[colBase + m];
    #pragma unroll
    for (int r = 0; r < 8; ++r) {       // C/D layout: VGPR r -> M = r + 8*hi, N = m
        int row = rowBase + r + 8 * hi;
        if (row < N)
            out[(long long)row * DFEAT + colBase + m] = fmaxf(c[r] + bb, 0.0f);
    }
}

extern "C" void kernel_launch(void* const* d_in, const int* in_sizes, int n_in,
                              void* d_out, int out_size, void* d_ws, size_t ws_size,
                              hipStream_t stream) {
    const float* x   = (const float*)d_in[0];
    const float* Wl1 = (const float*)d_in[1];
    const float* bl1 = (const float*)d_in[2];
    const float* Wr1 = (const float*)d_in[3];
    const float* Wl2 = (const float*)d_in[4];
    const float* bl2 = (const float*)d_in[5];
    const float* Wr2 = (const float*)d_in[6];
    const int*   ei  = (const int*)d_in[7];

    const int N = in_sizes[0] / DFEAT;
    const int E = in_sizes[7] / 2;
    const int* src = ei;        // edge_index[0]
    const int* dst = ei + E;    // edge_index[1]

    const int nblkScan = (N + SCAN_B - 1) / SCAN_B;

    // ---- workspace layout (agg first: 16B-aligned float4 traffic) ----
    float* agg     = (float*)d_ws;                       // N*D floats
    int*   cnt     = (int*)(agg + (long long)N * DFEAT); // N
    int*   offs    = cnt + N;                            // N+1
    int*   cursor  = offs + N + 1;                       // N
    int*   bsum    = cursor + N;                         // nblkScan
    int*   eSorted = bsum + nblkScan;                    // E
    float* out     = (float*)d_out;

    // ---- build CSR by destination (once; shared by both layers) ----
    sage_zero_int_kernel<<<1024, 256, 0, stream>>>(cnt, N);
    sage_hist_kernel<<<(E + 255) / 256, 256, 0, stream>>>(dst, cnt, E);
    sage_scan_block_kernel<<<nblkScan, SCAN_B, 0, stream>>>(cnt, offs, bsum, N);
    sage_scan_sums_kernel<<<1, 32, 0, stream>>>(bsum, nblkScan);
    sage_scan_add_kernel<<<nblkScan, SCAN_B, 0, stream>>>(offs, bsum, cursor, N, E);
    sage_fill_kernel<<<(E + 255) / 256, 256, 0, stream>>>(src, dst, cursor, eSorted, E);

    const int gemmBlocks = (N + 15) / 16;
    const int aggBlocks  = (N + 7) / 8;    // 8 waves (nodes) per 256-thread block

    // ---- layer 1 ----
    sage_aggregate_kernel<<<aggBlocks, 256, 0, stream>>>(x, offs, eSorted, agg, N);
    sage_gemm_kernel<<<gemmBlocks, 256, 0, stream>>>(agg, x, Wl1, bl1, Wr1, out, N);

    // ---- layer 2 ----
    sage_aggregate_kernel<<<aggBlocks, 256, 0, stream>>>(out, offs, eSorted, agg, N);
    sage_gemm_kernel<<<gemmBlocks, 256, 0, stream>>>(agg, out, Wl2, bl2, Wr2, out, N);
}